// RotationInvariantDistFea_23519240913431
// MI455X (gfx1250) — compile-verified
//
#include <hip/hip_runtime.h>

#define BATCH    2
#define NPTS     8192
#define KNN      16
#define APTS     4
#define TILE_PTS 1024   // candidate points staged in LDS per pass
#define FEAT_C   28     // 6 center-intra + 6 neighbor-intra + 16 inter

typedef __attribute__((ext_vector_type(2))) float    v2f;
typedef __attribute__((ext_vector_type(8))) float    v8f;
typedef __attribute__((ext_vector_type(4))) unsigned v4u;
typedef __attribute__((ext_vector_type(4))) int      v4i;
typedef __attribute__((ext_vector_type(8))) int      v8i;
typedef unsigned long long u64;

__device__ __forceinline__ float3 loadp(const float* __restrict__ xb, int i) {
  const float* p = xb + 3 * i;
  return make_float3(p[0], p[1], p[2]);
}

__device__ __forceinline__ float pdist(float3 a, float3 b) {
  float dx = a.x - b.x, dy = a.y - b.y, dz = a.z - b.z;
  return sqrtf(fmaxf(dx * dx + dy * dy + dz * dz, 0.0f));
}

__device__ __forceinline__ u64 umin64(u64 a, u64 b) { return a < b ? a : b; }
__device__ __forceinline__ u64 umax64(u64 a, u64 b) { return a > b ? a : b; }

// ---------------------------------------------------------------------------
// Tensor Data Mover: 1-row tile of `ndwords` 4-byte elements, global -> LDS.
// D# per CDNA5 ISA ch.8: group0 = {flags/count, lds_addr, global_addr, type=2},
// group1 = {data_size=4B, tensor_dim0, tile_dim0, stride}.
// ---------------------------------------------------------------------------
#if __has_builtin(__builtin_amdgcn_tensor_load_to_lds)
#define HAVE_TDM 1
__device__ __forceinline__ void tdm_load_1d(unsigned lds_off, const void* gptr,
                                            int ndwords) {
  unsigned long long ga = (unsigned long long)gptr;
  v4u g0;
  g0[0] = 1u;                                    // count=1, user descriptor
  g0[1] = lds_off;                               // lds_addr (byte offset)
  g0[2] = (unsigned)(ga & 0xffffffffu);          // global_addr[31:0]
  g0[3] = (unsigned)((ga >> 32) & 0x01ffffffu)   // global_addr[56:32]
        | 0x80000000u;                           // type=2 ("image") in [127:126]
  v8i g1;
  g1[0] = 0x00020000;                            // wg_mask=0, data_size=2 (4B)
  g1[1] = (ndwords & 0xffff) << 16;              // tensor_dim0[15:0]
  g1[2] = ((ndwords >> 16) & 0xffff) | (1 << 16);// tensor_dim0[31:16], tensor_dim1=1
  g1[3] = (ndwords & 0xffff) << 16;              // tile_dim0 = ndwords
  g1[4] = 1;                                     // tile_dim1 = 1, tile_dim2 = 0
  g1[5] = ndwords;                               // tensor_dim0_stride[31:0]
  g1[6] = 0;
  g1[7] = 0;
  v4i z4 = {0, 0, 0, 0};
#if defined(__clang_major__) && __clang_major__ >= 23
  v8i z8 = {0, 0, 0, 0, 0, 0, 0, 0};
  __builtin_amdgcn_tensor_load_to_lds(g0, g1, z4, z4, z8, 0);
#else
  __builtin_amdgcn_tensor_load_to_lds(g0, g1, z4, z4, 0);
#endif
}
#else
#define HAVE_TDM 0
#warning "gfx1250 __builtin_amdgcn_tensor_load_to_lds not available; using global-load staging fallback"
#endif

// ---------------------------------------------------------------------------
// Phase 1: brute-force kNN (k=16) via V_WMMA_F32_16X16X4_F32 distance tiles.
//   A row i = (-2x_i,-2y_i,-2z_i,1), B col j = (x_j,y_j,z_j,|p_j|^2),
//   C = |q_i|^2  =>  D = d2(i,j) in one WMMA per 16x16 tile.
// Candidates are DMA'd into LDS by the TDM, packed to float4, swept by all
// 8 waves; each wave transposes its two 16x16 D tiles through LDS and each
// lane keeps a sorted top-16 of packed u64 (distbits<<32 | idx) keys:
// branchless insert  new[s] = min(old[s], max(old[s-1], key)).
// ---------------------------------------------------------------------------
__global__ __launch_bounds__(256) void knn_topk_wmma(const float* __restrict__ xyz,
                                                     int* __restrict__ idx_out) {
  __shared__ float4 spts[TILE_PTS];                        // 16 KB packed stage
  __shared__ float  stile[8][32 * 16];                     // 16 KB d2 tiles
  __shared__ __align__(16) float sraw[TILE_PTS * 3];       // 12 KB TDM target

  const int  tid  = threadIdx.x;
  const int  lane = tid & 31;          // wave32
  const int  wave = tid >> 5;
  const int  bpb  = NPTS / 256;        // blocks per batch
  const int  b    = blockIdx.x / bpb;
  const int  rowBase = (blockIdx.x % bpb) * 256 + wave * 32;
  const bool hi   = lane >= 16;
  const int  l16  = lane & 15;

  const float* __restrict__ xb = xyz + (size_t)b * NPTS * 3;
  __builtin_prefetch(xb, 0, 1);        // global_prefetch_b8

  // A operands: lanes 0-15 carry K={0,1}, lanes 16-31 carry K={2,3} of row M=l16.
  const int rA0 = rowBase + l16;
  float3 p0 = loadp(xb, rA0);
  float3 p1 = loadp(xb, rA0 + 16);
  v2f a0, a1;
  a0.x = hi ? -2.0f * p0.z : -2.0f * p0.x;
  a0.y = hi ?  1.0f        : -2.0f * p0.y;
  a1.x = hi ? -2.0f * p1.z : -2.0f * p1.x;
  a1.y = hi ?  1.0f        : -2.0f * p1.y;

  // C operands: c[v] = |q|^2 of row M = v + (hi?8:0); second tile = +16.
  v8f c0, c1;
#pragma unroll
  for (int v = 0; v < 8; ++v) {
    int r0 = rowBase + v + (hi ? 8 : 0);
    float3 q0 = loadp(xb, r0);
    float3 q1 = loadp(xb, r0 + 16);
    c0[v] = q0.x * q0.x + q0.y * q0.y + q0.z * q0.z;
    c1[v] = q1.x * q1.x + q1.y * q1.y + q1.z * q1.z;
  }

  // Sorted ascending top-16 of packed keys: (bits(max(d2,0)) << 32) | idx.
  // Non-negative floats order as unsigned ints; idx in the low word makes
  // ties stable toward the smaller index (matches jax.lax.top_k).
  u64 best[KNN];
#pragma unroll
  for (int s = 0; s < KNN; ++s) best[s] = ~0ull;

  float* T = &stile[wave][0];

  for (int pass = 0; pass < NPTS / TILE_PTS; ++pass) {
#if HAVE_TDM
    if (tid == 0) {
      tdm_load_1d((unsigned)(uintptr_t)&sraw[0],
                  xb + (size_t)pass * TILE_PTS * 3, TILE_PTS * 3);
      __builtin_amdgcn_s_wait_tensorcnt(0);
    }
    __syncthreads();
    for (int i = tid; i < TILE_PTS; i += 256) {
      float x = sraw[3 * i], y = sraw[3 * i + 1], z = sraw[3 * i + 2];
      spts[i] = make_float4(x, y, z, x * x + y * y + z * z);
    }
#else
    for (int i = tid; i < TILE_PTS; i += 256) {
      int g = pass * TILE_PTS + i;
      float x = xb[3 * g], y = xb[3 * g + 1], z = xb[3 * g + 2];
      spts[i] = make_float4(x, y, z, x * x + y * y + z * z);
    }
#endif
    __syncthreads();

    for (int c = 0; c < TILE_PTS; c += 16) {
      float4 pc = spts[c + l16];
      v2f bb;                      // B col N=l16: lanes<16 -> (x,y); hi -> (z,|p|^2)
      bb.x = hi ? pc.z : pc.x;
      bb.y = hi ? pc.w : pc.y;

      v8f d0 = __builtin_amdgcn_wmma_f32_16x16x4_f32(false, a0, false, bb,
                                                     (short)0, c0, false, false);
      v8f d1 = __builtin_amdgcn_wmma_f32_16x16x4_f32(false, a1, false, bb,
                                                     (short)0, c1, false, false);

      // Transpose D through per-wave LDS tile (rows 0..15 = tile0, 16..31 = tile1).
#pragma unroll
      for (int v = 0; v < 8; ++v) {
        int tr = v + (hi ? 8 : 0);
        T[tr * 16 + l16]        = d0[v];
        T[(tr + 16) * 16 + l16] = d1[v];
      }
      asm volatile("s_wait_dscnt 0" ::: "memory"); // cross-lane LDS RAW inside wave

      // Lane owns query row rowBase+lane: fetch its 16 candidates as 4x b128.
      const float4* rp4 = (const float4*)(T + lane * 16);
      float4 q0 = rp4[0], q1 = rp4[1], q2 = rp4[2], q3 = rp4[3];
      float cand[16] = {q0.x, q0.y, q0.z, q0.w, q1.x, q1.y, q1.z, q1.w,
                        q2.x, q2.y, q2.z, q2.w, q3.x, q3.y, q3.z, q3.w};
      const int cbase = pass * TILE_PTS + c;
#pragma unroll
      for (int j = 0; j < 16; ++j) {
        float dd  = fmaxf(cand[j], 0.0f);
        u64   key = ((u64)__float_as_uint(dd) << 32) | (unsigned)(cbase + j);
        if (key < best[KNN - 1]) {   // cheap reject dominates after warm-up
          // Branchless sorted insert; s descending reads pre-update values.
#pragma unroll
          for (int s = KNN - 1; s > 0; --s)
            best[s] = umin64(best[s], umax64(best[s - 1], key));
          best[0] = umin64(best[0], key);
        }
      }
    }
    __syncthreads();
  }

  const int row = rowBase + lane;
  int* op = idx_out + ((size_t)b * NPTS + row) * KNN;
#pragma unroll
  for (int s = 0; s < KNN; ++s) op[s] = (int)(unsigned)(best[s] & 0xffffffffu);
}

// ---------------------------------------------------------------------------
// Phase 2: rotation-invariant features from xyz + neighbor indices only.
// One thread per (b, n, k): 10 point gathers, 28 sqrt distances.
// ---------------------------------------------------------------------------
__global__ __launch_bounds__(256) void rifeat_kernel(const float* __restrict__ xyz,
                                                     const int* __restrict__ idx_in,
                                                     float* __restrict__ feat) {
  int gid = blockIdx.x * 256 + threadIdx.x;
  int b   = gid / (NPTS * KNN);
  int rem = gid - b * (NPTS * KNN);
  int n   = rem / KNN;
  int k   = rem - n * KNN;

  const float* __restrict__ xb = xyz + (size_t)b * NPTS * 3;
  const int* __restrict__ ib   = idx_in + (size_t)b * NPTS * KNN;
  __builtin_prefetch(xb, 0, 1);

  int ca[APTS];
#pragma unroll
  for (int a = 0; a < APTS; ++a) ca[a] = ib[n * KNN + a];
  int nb = ib[n * KNN + k];
  int na[APTS];
#pragma unroll
  for (int a = 0; a < APTS; ++a) na[a] = ib[nb * KNN + a];

  float3 Pn  = loadp(xb, n);
  float3 Pnb = loadp(xb, nb);
  float3 Ca[APTS], Na[APTS];
#pragma unroll
  for (int a = 0; a < APTS; ++a) { Ca[a] = loadp(xb, ca[a]); Na[a] = loadp(xb, na[a]); }

  float* o = feat + (((size_t)b * NPTS + n) * KNN + k) * FEAT_C;
  // center intra: [d(n,a1), d(n,a2), d(n,a3), d(a1,a2), d(a1,a3), d(a2,a3)]
  o[0] = pdist(Pn, Ca[1]);    o[1] = pdist(Pn, Ca[2]);    o[2] = pdist(Pn, Ca[3]);
  o[3] = pdist(Ca[1], Ca[2]); o[4] = pdist(Ca[1], Ca[3]); o[5] = pdist(Ca[2], Ca[3]);
  // neighbor intra
  o[6] = pdist(Pnb, Na[1]);   o[7] = pdist(Pnb, Na[2]);   o[8] = pdist(Pnb, Na[3]);
  o[9] = pdist(Na[1], Na[2]); o[10] = pdist(Na[1], Na[3]); o[11] = pdist(Na[2], Na[3]);
  // inter: dist(center_anchor[pi], neighbor_anchor[pj]), pi-major
#pragma unroll
  for (int pi = 0; pi < APTS; ++pi)
#pragma unroll
    for (int pj = 0; pj < APTS; ++pj)
      o[12 + pi * APTS + pj] = pdist(Ca[pi], Na[pj]);
}

extern "C" void kernel_launch(void* const* d_in, const int* in_sizes, int n_in,
                              void* d_out, int out_size, void* d_ws, size_t ws_size,
                              hipStream_t stream) {
  (void)in_sizes; (void)n_in; (void)out_size; (void)d_ws; (void)ws_size;
  const float* xyz = (const float*)d_in[0];

  float* feat = (float*)d_out;
  // int32 index output lives after the float feature block (return-order concat)
  int* idx_out = (int*)d_out + (size_t)BATCH * NPTS * KNN * FEAT_C;

  knn_topk_wmma<<<BATCH * (NPTS / 256), 256, 0, stream>>>(xyz, idx_out);
  rifeat_kernel<<<(BATCH * NPTS * KNN) / 256, 256, 0, stream>>>(xyz, idx_out, feat);
}